// SoftCrossEntropyLossByNegSampling_72095321030873
// MI455X (gfx1250) — compile-verified
//
#include <hip/hip_runtime.h>
#include <hip/hip_bf16.h>
#include <stdint.h>

// SoftCrossEntropyLossByNegSampling for MI455X (gfx1250).
// Gather-bound: ~58MB HBM traffic, ~4 MFLOP -> WMMA inapplicable.
// CDNA5 paths used: global_load_async_to_lds_b128 (ASYNCcnt),
// s_wait_asynccnt, non-temporal gather hints, wave32 shuffle reduction.

#define POSN 20          // P: positives per sample (fixed by reference)
#define BLOCK 256        // 8 wave32s per row

// 16-byte async global -> LDS copy (one chunk per active lane).
// vdst holds the LDS byte address, vaddr the 64-bit global address, GV mode.
__device__ __forceinline__ void async_copy_b128(void* lds_dst, const void* gsrc) {
    // Generic pointers to LDS are {aperture_hi32, lds_offset32}; low 32 bits
    // are exactly the LDS address the vdst operand needs (hardware adds the LDS base).
    uint32_t lds_off = (uint32_t)(uintptr_t)lds_dst;
    asm volatile("global_load_async_to_lds_b128 %0, %1, off"
                 :
                 : "v"(lds_off), "v"(gsrc)
                 : "memory");
}

__device__ __forceinline__ void wait_async_all() {
    asm volatile("s_wait_asynccnt 0" ::: "memory");
}

// Numerically stable log_sigmoid(x) = min(x,0) - log1p(exp(-|x|))
__device__ __forceinline__ float log_sigmoid(float x) {
    float ax = __builtin_fabsf(x);
    return fminf(x, 0.0f) - __logf(1.0f + __expf(-ax));
}

__global__ void __launch_bounds__(BLOCK)
nce_row_kernel(const float* __restrict__ output,   // [B, V]
               const int*   __restrict__ keys,     // [B, P]
               const float* __restrict__ vals,     // [B, P]
               const int*   __restrict__ negs,     // [B, N]
               float*       __restrict__ partial,  // [B]
               int V, int N) {
    const int b = blockIdx.x;
    const int t = threadIdx.x;

    __shared__ int   s_keys[32];          // 20 used, padded
    __shared__ float s_vals[32];
    __shared__ int   s_negs[512];         // N = 200 used
    __shared__ float s_red[3 * (BLOCK / 32)];

    // ---- Stage index/weight vectors into LDS with async tensor-path copies.
    // keys: 80B = 5x16B chunks, vals: 5 chunks, negs: N ints -> N/4 chunks.
    const int negChunks = N >> 2;         // groups of 4 ints = 16B
    if (t < 5) {
        async_copy_b128(&s_keys[t * 4], keys + (size_t)b * POSN + t * 4);
    } else if (t < 10) {
        const int c = t - 5;
        async_copy_b128(&s_vals[c * 4], vals + (size_t)b * POSN + c * 4);
    } else if (t < 10 + negChunks) {
        const int c = t - 10;
        async_copy_b128(&s_negs[c * 4], negs + (size_t)b * N + c * 4);
    }
    wait_async_all();       // issuing wave drains its ASYNCcnt
    __syncthreads();        // LDS now visible to whole workgroup

    // ---- Gather + log-sigmoid. One element per thread (220 active).
    const float* __restrict__ row = output + (size_t)b * V;
    float a  = 0.0f;   // sum v * log_sigmoid(pos)
    float vs = 0.0f;   // sum v            (defer normalization: s1 = a/vs)
    float c  = 0.0f;   // sum log_sigmoid(-neg)

    if (t < POSN) {
        const int   k = s_keys[t];
        const float v = s_vals[t];
        // One-touch random read of an 819MB matrix: non-temporal so the
        // 192MB L2 is not thrashed.
        const float x = __builtin_nontemporal_load(row + k);
        a  = v * log_sigmoid(x);
        vs = v;
    } else if (t < POSN + N) {
        const int k = s_negs[t - POSN];
        const float x = __builtin_nontemporal_load(row + k);
        c = log_sigmoid(-x);
    }

    // ---- wave32 shuffle reduction, then cross-wave via LDS.
    #pragma unroll
    for (int off = 16; off > 0; off >>= 1) {
        a  += __shfl_down(a,  off, 32);
        vs += __shfl_down(vs, off, 32);
        c  += __shfl_down(c,  off, 32);
    }
    const int wave = t >> 5;
    const int lane = t & 31;
    if (lane == 0) {
        s_red[wave * 3 + 0] = a;
        s_red[wave * 3 + 1] = vs;
        s_red[wave * 3 + 2] = c;
    }
    __syncthreads();
    if (t == 0) {
        float A = 0.0f, Vs = 0.0f, C = 0.0f;
        #pragma unroll
        for (int w = 0; w < BLOCK / 32; ++w) {
            A  += s_red[w * 3 + 0];
            Vs += s_red[w * 3 + 1];
            C  += s_red[w * 3 + 2];
        }
        partial[b] = A / Vs + C;   // s1 + s2 for this row
    }
}

// Deterministic fixed-order mean of B partials (no float atomics -> bitwise
// reproducible across graph replays).
__global__ void __launch_bounds__(BLOCK)
reduce_mean_kernel(const float* __restrict__ partial, float* __restrict__ out, int B) {
    __shared__ float s[BLOCK];
    const int t = threadIdx.x;
    float acc = 0.0f;
    for (int i = t; i < B; i += BLOCK) acc += partial[i];  // fixed per-thread order
    s[t] = acc;
    __syncthreads();
    #pragma unroll
    for (int off = BLOCK / 2; off > 0; off >>= 1) {
        if (t < off) s[t] += s[t + off];
        __syncthreads();
    }
    if (t == 0) out[0] = s[0] / (float)B;
}

extern "C" void kernel_launch(void* const* d_in, const int* in_sizes, int n_in,
                              void* d_out, int out_size, void* d_ws, size_t ws_size,
                              hipStream_t stream) {
    const float* output = (const float*)d_in[0];   // [B, V] f32
    const int*   keys   = (const int*)  d_in[1];   // [B, P] int
    const float* vals   = (const float*)d_in[2];   // [B, P] f32
    const int*   negs   = (const int*)  d_in[3];   // [B, N] int

    const int B = in_sizes[2] / POSN;              // target_values is [B, P]
    const int V = in_sizes[0] / B;
    const int N = in_sizes[3] / B;                 // P*K = 200

    float* partial = (float*)d_ws;                 // B floats of scratch

    nce_row_kernel<<<B, BLOCK, 0, stream>>>(output, keys, vals, negs, partial, V, N);
    reduce_mean_kernel<<<1, BLOCK, 0, stream>>>(partial, (float*)d_out, B);
}